// CCE_67190468378875
// MI455X (gfx1250) — compile-verified
//
#include <hip/hip_runtime.h>

// Problem constants (match reference)
#define B_N   4096
#define C_N   200
#define P_N   32
#define F_N   512
#define CP_N  6400          // C_N * P_N
#define ALPHA 5.0f
#define EPS_F 1e-8f

#define LDA        516      // padded LDS row stride in floats (516 % 64 == 4 -> conflict-free)
#define NSPLIT     4        // column splits across workgroups
#define TILES_SPL  100      // (CP_N / NSPLIT) / 16
#define ROWS_WG    64       // batch rows per workgroup (4 WMMA row tiles)
#define INF_F      __builtin_inff()

// Workspace layout (bytes)
#define WS_P2_OFF   0
#define WS_PART_OFF 32768                          // NSPLIT * B_N * 16 bytes
#define WS_TSUM_OFF (WS_PART_OFF + NSPLIT * B_N * 16)
#define WS_OSUM_OFF (WS_TSUM_OFF + B_N * 4)

typedef __attribute__((ext_vector_type(2))) float v2f;
typedef __attribute__((ext_vector_type(8))) float v8f;

// Branchless (min, argmin) combine with first-index tie-break.
__device__ __forceinline__ void argmin_combine(float& v, int& i, float v2, int i2) {
  const bool take = (v2 < v) | ((v2 == v) & (i2 < i));
  v = take ? v2 : v;
  i = take ? i2 : i;
}

// ---------------- Phase 1: prototype squared norms p2[cp] ----------------
__global__ __launch_bounds__(256) void k_p2(const float* __restrict__ clus,
                                            float* __restrict__ p2f) {
  const int lane = threadIdx.x & 31;
  const int wave = threadIdx.x >> 5;
  const int cp = blockIdx.x * 8 + wave;
  if (cp >= CP_N) return;
  const float* p = clus + (size_t)cp * F_N;
  float s = 0.f;
  #pragma unroll 4
  for (int k = lane; k < F_N; k += 32) { const float v = p[k]; s += v * v; }
  #pragma unroll
  for (int m = 16; m; m >>= 1) s += __shfl_xor(s, m, 32);
  if (lane == 0) p2f[cp] = s;
}

// ---------------- Phase 2: fused WMMA distance GEMM + argmin ----------------
// d2(b,cp) argmin-equivalent score: p2[cp] - 2*dot(x[b],clus[cp])
// A (staged in LDS) = -2 * outputs ; B = clusters^T streamed from L2.
__global__ __launch_bounds__(256) void k_dist(const float* __restrict__ xout,
                                              const float* __restrict__ clus,
                                              const int*   __restrict__ tcls,
                                              const float* __restrict__ p2f,
                                              float* __restrict__ part) {
  extern __shared__ float lds[];
  float* redm = lds + ROWS_WG * LDA;           // [8 waves][64 rows][4 floats]

  const int tid  = threadIdx.x;
  const int lane = tid & 31;
  const int wave = tid >> 5;
  const int rg    = blockIdx.x >> 2;           // row group 0..63
  const int split = blockIdx.x & 3;            // column split 0..3
  const int rowBase = rg * ROWS_WG;

  // Stage A tile: 64 rows x 512 cols, scaled by -2, padded stride LDA
  for (int i = tid; i < ROWS_WG * (F_N / 4); i += 256) {
    const int r  = i >> 7;                     // F_N/4 == 128
    const int c4 = (i & 127) << 2;
    const float4 v = *(const float4*)(xout + (size_t)(rowBase + r) * F_N + c4);
    float4 w; w.x = -2.f * v.x; w.y = -2.f * v.y; w.z = -2.f * v.z; w.w = -2.f * v.w;
    *(float4*)&lds[r * LDA + c4] = w;
  }
  __syncthreads();

  const int hi    = lane >> 4;                 // 0: rows j, 1: rows j+8 (C/D layout)
  const int hi8   = hi << 3;
  const int khalf = hi << 1;                   // A/B fragment K sub-offset (0 or 2)
  const int nlo   = lane & 15;

  // Target class per (row tile, accumulator element)
  int tgt[4][8];
  #pragma unroll
  for (int r = 0; r < 4; ++r)
    #pragma unroll
    for (int j = 0; j < 8; ++j)
      tgt[r][j] = tcls[rowBase + r * 16 + j + hi8];

  float mt[4][8], mo[4][8];
  int   it[4][8], io[4][8];
  #pragma unroll
  for (int r = 0; r < 4; ++r)
    #pragma unroll
    for (int j = 0; j < 8; ++j) {
      mt[r][j] = INF_F; mo[r][j] = INF_F; it[r][j] = 0; io[r][j] = 0;
    }

  const float* aptr = lds + nlo * LDA + khalf;

  for (int t = wave; t < TILES_SPL; t += 8) {
    const int colBase = split * (CP_N / NSPLIT) + t * 16;
    const int cp  = colBase + nlo;             // this lane's prototype column
    const float p2l = p2f[cp];
    const int cls = cp >> 5;                   // class of this column
    const float* bptr = clus + (size_t)cp * F_N + khalf;

    v8f acc[4];
    {
      const v8f z = {};
      #pragma unroll
      for (int r = 0; r < 4; ++r) acc[r] = z;
    }

    #pragma unroll 4
    for (int k = 0; k < F_N; k += 4) {
      const v2f bf = *(const v2f*)(bptr + k);                 // global_load_b64
      const v2f a0 = *(const v2f*)(aptr + k);                 // ds_load_b64, conflict-free
      const v2f a1 = *(const v2f*)(aptr + 16 * LDA + k);
      const v2f a2 = *(const v2f*)(aptr + 32 * LDA + k);
      const v2f a3 = *(const v2f*)(aptr + 48 * LDA + k);
      acc[0] = __builtin_amdgcn_wmma_f32_16x16x4_f32(false, a0, false, bf, (short)0, acc[0], false, false);
      acc[1] = __builtin_amdgcn_wmma_f32_16x16x4_f32(false, a1, false, bf, (short)0, acc[1], false, false);
      acc[2] = __builtin_amdgcn_wmma_f32_16x16x4_f32(false, a2, false, bf, (short)0, acc[2], false, false);
      acc[3] = __builtin_amdgcn_wmma_f32_16x16x4_f32(false, a3, false, bf, (short)0, acc[3], false, false);
    }

    // Fold 16x16 tiles into per-lane running (min, argmin) — fully branchless:
    // route candidate to +inf for the tracker it doesn't belong to.
    // cp strictly increases per lane across tiles -> strict '<' keeps first index.
    #pragma unroll
    for (int r = 0; r < 4; ++r)
      #pragma unroll
      for (int j = 0; j < 8; ++j) {
        const float v   = acc[r][j] + p2l;
        const bool  isT = (cls == tgt[r][j]);
        const float vt_c = isT ? v : INF_F;
        const float vo_c = isT ? INF_F : v;
        const bool bt = vt_c < mt[r][j];
        mt[r][j] = bt ? vt_c : mt[r][j];
        it[r][j] = bt ? cp   : it[r][j];
        const bool bo = vo_c < mo[r][j];
        mo[r][j] = bo ? vo_c : mo[r][j];
        io[r][j] = bo ? cp   : io[r][j];
      }
  }

  // Cross-lane reduction within each 16-lane half (columns live across lanes)
  #pragma unroll
  for (int r = 0; r < 4; ++r)
    #pragma unroll
    for (int j = 0; j < 8; ++j) {
      float vt = mt[r][j]; int jt = it[r][j];
      float vo = mo[r][j]; int jo = io[r][j];
      #pragma unroll
      for (int m = 8; m; m >>= 1) {
        const float vt2 = __shfl_xor(vt, m, 32);
        const int   jt2 = __shfl_xor(jt, m, 32);
        argmin_combine(vt, jt, vt2, jt2);
        const float vo2 = __shfl_xor(vo, m, 32);
        const int   jo2 = __shfl_xor(jo, m, 32);
        argmin_combine(vo, jo, vo2, jo2);
      }
      if (nlo == 0) {
        const int rowLocal = r * 16 + j + hi8;
        float* q = &redm[(wave * ROWS_WG + rowLocal) * 4];
        q[0] = vt; ((int*)q)[1] = jt; q[2] = vo; ((int*)q)[3] = jo;
      }
    }
  __syncthreads();

  // Cross-wave reduction (8 waves), write per-(row, split) partials
  if (tid < ROWS_WG) {
    float vt = INF_F; int jt = 0; float vo = INF_F; int jo = 0;
    #pragma unroll
    for (int w = 0; w < 8; ++w) {
      const float* q = &redm[(w * ROWS_WG + tid) * 4];
      argmin_combine(vt, jt, q[0], ((const int*)q)[1]);
      argmin_combine(vo, jo, q[2], ((const int*)q)[3]);
    }
    float* q = part + (size_t)(split * B_N + rowBase + tid) * 4;
    q[0] = vt; ((int*)q)[1] = jt; q[2] = vo; ((int*)q)[3] = jo;
  }
}

// ---------------- Phase 3: per-row gather + squared-distance sums ----------------
__global__ __launch_bounds__(256) void k_loss(const float* __restrict__ xout,
                                              const float* __restrict__ clus,
                                              const float* __restrict__ part,
                                              float* __restrict__ tsum,
                                              float* __restrict__ osum) {
  const int lane = threadIdx.x & 31;
  const int wave = threadIdx.x >> 5;
  const int b = blockIdx.x * 8 + wave;
  if (b >= B_N) return;

  float vt = INF_F; int jt = 0; float vo = INF_F; int jo = 0;
  #pragma unroll
  for (int s = 0; s < NSPLIT; ++s) {
    const float* q = part + (size_t)(s * B_N + b) * 4;
    argmin_combine(vt, jt, q[0], ((const int*)q)[1]);
    argmin_combine(vo, jo, q[2], ((const int*)q)[3]);
  }

  const float* x  = xout + (size_t)b * F_N;
  const float* pa = clus + (size_t)jt * F_N;   // actual (target-class) prototype
  const float* pw = clus + (size_t)jo * F_N;   // wrong-class prototype
  float s1 = 0.f, s2 = 0.f;
  #pragma unroll 4
  for (int k = lane; k < F_N; k += 32) {
    const float xv = x[k];
    const float d1 = xv - pa[k]; s1 += d1 * d1;
    const float d2 = xv - pw[k]; s2 += d2 * d2;
  }
  #pragma unroll
  for (int m = 16; m; m >>= 1) { s1 += __shfl_xor(s1, m, 32); s2 += __shfl_xor(s2, m, 32); }
  if (lane == 0) { tsum[b] = s1; osum[b] = s2; }
}

// ---------------- Phase 4: deterministic final reduction + scalar ----------------
__global__ __launch_bounds__(256) void k_final(const float* __restrict__ tsum,
                                               const float* __restrict__ osum,
                                               float* __restrict__ outp) {
  __shared__ float st[256], so[256];
  const int tid = threadIdx.x;
  float a = 0.f, b = 0.f;
  for (int i = tid; i < B_N; i += 256) { a += tsum[i]; b += osum[i]; }
  st[tid] = a; so[tid] = b;
  __syncthreads();
  for (int s = 128; s; s >>= 1) {
    if (tid < s) { st[tid] += st[tid + s]; so[tid] += so[tid + s]; }
    __syncthreads();
  }
  if (tid == 0) {
    const float denom = (float)B_N * (float)F_N;
    const float tl = st[0] / denom;
    const float ol = so[0] / denom;
    outp[0] = (1.0f - ALPHA) * tl + ALPHA * (1.0f / (ol + EPS_F));
  }
}

extern "C" void kernel_launch(void* const* d_in, const int* in_sizes, int n_in,
                              void* d_out, int out_size, void* d_ws, size_t ws_size,
                              hipStream_t stream) {
  const float* xout = (const float*)d_in[0];   // outputs [B,F] fp32
  const float* clus = (const float*)d_in[1];   // clusters [C,P,F] fp32
  const int*   tcls = (const int*)d_in[2];     // target_classes [B] int32
  float* out = (float*)d_out;

  char* w = (char*)d_ws;
  float* p2f  = (float*)(w + WS_P2_OFF);
  float* part = (float*)(w + WS_PART_OFF);
  float* tsum = (float*)(w + WS_TSUM_OFF);
  float* osum = (float*)(w + WS_OSUM_OFF);

  k_p2<<<CP_N / 8, 256, 0, stream>>>(clus, p2f);

  const size_t shmem = (size_t)(ROWS_WG * LDA + 8 * ROWS_WG * 4) * sizeof(float);
  k_dist<<<(B_N / ROWS_WG) * NSPLIT, 256, shmem, stream>>>(xout, clus, tcls, p2f, part);

  k_loss<<<B_N / 8, 256, 0, stream>>>(xout, clus, part, tsum, osum);
  k_final<<<1, 256, 0, stream>>>(tsum, osum, out);
}